// GeneratorTest_15891378995362
// MI455X (gfx1250) — compile-verified
//
#include <hip/hip_runtime.h>
#include <math.h>

// ---------------------------------------------------------------------------
// Adaptive softmax loss: fused f32-WMMA GEMM + online log-sum-exp, gfx1250.
// Split-V partial-LSE blocks + merge pass; RT row-tiles share each B fragment.
// Target / cluster-column logits extracted by a separate cheap gather-dot.
// ---------------------------------------------------------------------------

typedef __attribute__((ext_vector_type(2))) float v2f;
typedef __attribute__((ext_vector_type(4))) float v4f;
typedef __attribute__((ext_vector_type(8))) float v8f;

#define NH        1024
#define VS        50257
#define CUT0      1675
#define CUT1      5025
#define HEAD_SIZE 1677
#define D0        256
#define D1        64
#define SZ0       3350
#define SZ1       45232
#define NTOK      4096
#define HSTRIDE   (D0 + D1)   // 320
#define MAXCHUNKS 16

__device__ __forceinline__ void lse_merge2(float& M, float& S, float om, float os) {
    float nm = fmaxf(M, om);
    // guard: both -inf would give exp(nan); keep S = 0 in that case
    S = (nm == -__builtin_inff()) ? 0.f
                                  : (S * __expf(M - nm) + os * __expf(om - nm));
    M = nm;
}

// ---------------------------------------------------------------------------
// Target index preparation (reference's where/clip semantics).
// ---------------------------------------------------------------------------
__global__ void prep_kernel(const int* __restrict__ y,
                            int* __restrict__ tgt_head,
                            int* __restrict__ tgt0,
                            int* __restrict__ tgt1) {
    int n = blockIdx.x * blockDim.x + threadIdx.x;
    if (n >= NTOK) return;
    int yy = y[n];
    tgt_head[n] = (yy < CUT0) ? yy : 0;
    int r0 = yy - CUT0; r0 = r0 < 0 ? 0 : (r0 > SZ0 - 1 ? SZ0 - 1 : r0);
    int r1 = yy - CUT1; r1 = r1 < 0 ? 0 : (r1 > SZ1 - 1 ? SZ1 - 1 : r1);
    tgt0[n] = r0;
    tgt1[n] = r1;
}

// ---------------------------------------------------------------------------
// Per-token gathered dot: out[n*stride + off] = A[n] . W[col(n)] (+ bias[col])
// col(n) = idx[n] (if idx) else fixed_col. One wave per token; negligible cost
// (<=0.1% of total FLOPs) — removes all extraction logic from the hot GEMM.
// ---------------------------------------------------------------------------
__launch_bounds__(256)
__global__ void gather_dot_kernel(const float* __restrict__ A, int lda, int a_col_off,
                                  const float* __restrict__ W,
                                  const float* __restrict__ bias, int K,
                                  const int* __restrict__ idx, int fixed_col,
                                  float* __restrict__ out, int out_stride, int out_off) {
    const int token = blockIdx.x * 8 + (threadIdx.x >> 5);
    const int lane  = threadIdx.x & 31;
    const int col   = idx ? idx[token] : fixed_col;
    const float* a = A + (size_t)token * lda + a_col_off;
    const float* w = W + (size_t)col * K;
    float sum = 0.f;
    for (int k = lane; k < K; k += 32) sum += a[k] * w[k];
#pragma unroll
    for (int off = 16; off >= 1; off >>= 1) sum += __shfl_xor(sum, off, 32);
    if (lane == 0)
        out[(size_t)token * out_stride + out_off] = sum + (bias ? bias[col] : 0.f);
}

// ---------------------------------------------------------------------------
// Partial fused GEMM + LSE over one V-chunk.
//   logits[16*RT rows x chunk] = A[16*RT x K] @ W[chunk x K]^T (+ bias)
// Block = 128 thr = 4 wave32; each wave owns one 16-col tile at a time; RT
// row-tiles share every streamed B fragment (RT independent WMMA chains).
// Emits per-row partial (max, sumexp) for this chunk.
// ---------------------------------------------------------------------------
template <int RT>
__launch_bounds__(128)
__global__ void fused_lse_partial(const float* __restrict__ A, int lda, int a_col_off,
                                  const float* __restrict__ W,
                                  const float* __restrict__ bias,
                                  int K, int V, int tiles_per_chunk,
                                  float* __restrict__ m_part,     // [chunks][NTOK]
                                  float* __restrict__ s_part) {   // [chunks][NTOK]
    extern __shared__ float smem[];                  // (16*RT) x (K+4) A-tile
    __shared__ float m_ws[4 * 16 * RT];
    __shared__ float s_ws[4 * 16 * RT];

    const int tid   = threadIdx.x;
    const int lane  = tid & 31;
    const int wv    = tid >> 5;                      // wave 0..3
    const int half  = lane >> 4;
    const int l16   = lane & 15;
    const int row0  = blockIdx.x * 16 * RT;
    const int chunk = blockIdx.y;
    const int ldpad = K + 4;

    // ---- stage A tile (16*RT x K) into LDS with 16B vector loads ----
    const int k4 = K >> 2;
    for (int idx = tid; idx < 16 * RT * k4; idx += blockDim.x) {
        int r = idx / k4, c = idx - r * k4;
        v4f val = *(const v4f*)(A + (size_t)(row0 + r) * lda + a_col_off + 4 * c);
        *(v4f*)(smem + r * ldpad + 4 * c) = val;
    }
    __syncthreads();

    float m[RT][8], s[RT][8];
#pragma unroll
    for (int rt = 0; rt < RT; ++rt)
#pragma unroll
        for (int v = 0; v < 8; ++v) { m[rt][v] = -__builtin_inff(); s[rt][v] = 0.f; }

    // A fragment bases: row = rt*16 + l16, K-pair starts at 2*half
    const float* apr[RT];
#pragma unroll
    for (int rt = 0; rt < RT; ++rt)
        apr[rt] = smem + (rt * 16 + l16) * ldpad + 2 * half;

    const int ntiles = (V + 15) >> 4;
    const int tstart = chunk * tiles_per_chunk;
    const int tend_  = tstart + tiles_per_chunk;
    const int tend   = tend_ < ntiles ? tend_ : ntiles;

    for (int t = tstart + wv; t < tend; t += 4) {
        const int col  = (t << 4) + l16;
        const bool cvalid = col < V;
        const int  colc = cvalid ? col : (V - 1);
        const float* wp = W + (size_t)colc * K + 2 * half;

        if (t + 4 < tend) {
            int pcol = ((t + 4) << 4) + l16;
            if (pcol < V) __builtin_prefetch(W + (size_t)pcol * K, 0, 1);
        }

        v8f acc[RT];
#pragma unroll
        for (int rt = 0; rt < RT; ++rt)
            acc[rt] = (v8f){0.f, 0.f, 0.f, 0.f, 0.f, 0.f, 0.f, 0.f};

#pragma unroll 2
        for (int k0 = 0; k0 < K; k0 += 16) {
            // 4 independent B-fragment loads in flight (distinct registers)
            v2f b0 = *(const v2f*)(wp + k0);
            v2f b1 = *(const v2f*)(wp + k0 + 4);
            v2f b2 = *(const v2f*)(wp + k0 + 8);
            v2f b3 = *(const v2f*)(wp + k0 + 12);
#pragma unroll
            for (int rt = 0; rt < RT; ++rt) {
                v2f a0 = *(const v2f*)(apr[rt] + k0);
                v2f a1 = *(const v2f*)(apr[rt] + k0 + 4);
                v2f a2 = *(const v2f*)(apr[rt] + k0 + 8);
                v2f a3 = *(const v2f*)(apr[rt] + k0 + 12);
                acc[rt] = __builtin_amdgcn_wmma_f32_16x16x4_f32(
                    false, a0, false, b0, (short)0, acc[rt], false, false);
                acc[rt] = __builtin_amdgcn_wmma_f32_16x16x4_f32(
                    false, a1, false, b1, (short)0, acc[rt], false, false);
                acc[rt] = __builtin_amdgcn_wmma_f32_16x16x4_f32(
                    false, a2, false, b2, (short)0, acc[rt], false, false);
                acc[rt] = __builtin_amdgcn_wmma_f32_16x16x4_f32(
                    false, a3, false, b3, (short)0, acc[rt], false, false);
            }
        }

        if (cvalid) {
            const float bv = bias ? bias[col] : 0.f;
#pragma unroll
            for (int rt = 0; rt < RT; ++rt) {
#pragma unroll
                for (int v = 0; v < 8; ++v) {
                    float logit = acc[rt][v] + bv;
                    if (logit > m[rt][v]) {           // online softmax update
                        s[rt][v] = s[rt][v] * __expf(m[rt][v] - logit) + 1.0f;
                        m[rt][v] = logit;
                    } else {
                        s[rt][v] += __expf(logit - m[rt][v]);
                    }
                }
            }
        }
    }

    // ---- reduce (m,s) across the 16 lanes of each half-wave ----
#pragma unroll
    for (int rt = 0; rt < RT; ++rt) {
#pragma unroll
        for (int v = 0; v < 8; ++v) {
#pragma unroll
            for (int off = 8; off >= 1; off >>= 1) {
                float om = __shfl_xor(m[rt][v], off, 32);
                float os = __shfl_xor(s[rt][v], off, 32);
                lse_merge2(m[rt][v], s[rt][v], om, os);
            }
        }
    }
    if (l16 == 0) {
#pragma unroll
        for (int rt = 0; rt < RT; ++rt)
#pragma unroll
            for (int v = 0; v < 8; ++v) {
                int r = rt * 16 + v + 8 * half;
                m_ws[wv * (16 * RT) + r] = m[rt][v];
                s_ws[wv * (16 * RT) + r] = s[rt][v];
            }
    }
    __syncthreads();

    if (tid < 16 * RT) {
        float M = m_ws[tid], S = s_ws[tid];
#pragma unroll
        for (int w = 1; w < 4; ++w)
            lse_merge2(M, S, m_ws[w * (16 * RT) + tid], s_ws[w * (16 * RT) + tid]);
        size_t o = (size_t)chunk * NTOK + row0 + tid;
        m_part[o] = M;
        s_part[o] = S;
    }
}

// ---------------------------------------------------------------------------
// Merge per-chunk partial (max,sumexp) -> final target log-prob(s).
// ---------------------------------------------------------------------------
__global__ void lse_merge_kernel(const float* __restrict__ m_part,
                                 const float* __restrict__ s_part,
                                 int nchunks,
                                 const float* __restrict__ tgt_logit,
                                 const float* __restrict__ cl_logit,
                                 float* __restrict__ out_lp,
                                 float* __restrict__ out_cl) {
    int n = blockIdx.x * blockDim.x + threadIdx.x;
    if (n >= NTOK) return;
    float M = -__builtin_inff(), S = 0.f;
    for (int c = 0; c < nchunks; ++c)
        lse_merge2(M, S, m_part[(size_t)c * NTOK + n], s_part[(size_t)c * NTOK + n]);
    float lse = M + __logf(S);
    out_lp[n] = tgt_logit[n] - lse;
    if (out_cl) {
        out_cl[2 * n + 0] = cl_logit[2 * n + 0] - lse;
        out_cl[2 * n + 1] = cl_logit[2 * n + 1] - lse;
    }
}

// ---------------------------------------------------------------------------
// Projection GEMM: H[4096 x 320] = x @ [proj0; proj1]^T  (K = 1024)
// ---------------------------------------------------------------------------
__launch_bounds__(128)
__global__ void proj_kernel(const float* __restrict__ A,
                            const float* __restrict__ P0,
                            const float* __restrict__ P1,
                            float* __restrict__ H) {
    extern __shared__ float smem[];                  // 16 x (NH+4)
    const int tid  = threadIdx.x;
    const int lane = tid & 31;
    const int wv   = tid >> 5;
    const int half = lane >> 4;
    const int l16  = lane & 15;
    const int row0 = blockIdx.x * 16;
    const int ldpad = NH + 4;

    const int k4 = NH >> 2;
    for (int idx = tid; idx < 16 * k4; idx += blockDim.x) {
        int r = idx / k4, c = idx - r * k4;
        v4f val = *(const v4f*)(A + (size_t)(row0 + r) * NH + 4 * c);
        *(v4f*)(smem + r * ldpad + 4 * c) = val;
    }
    __syncthreads();

    const float* ap = smem + l16 * ldpad + 2 * half;

    for (int t = wv; t < HSTRIDE / 16; t += 4) {
        const int col = (t << 4) + l16;
        const float* wrow = (col < D0) ? (P0 + (size_t)col * NH)
                                       : (P1 + (size_t)(col - D0) * NH);
        const float* wp = wrow + 2 * half;

        v8f acc = {0.f, 0.f, 0.f, 0.f, 0.f, 0.f, 0.f, 0.f};
#pragma unroll 2
        for (int k0 = 0; k0 < NH; k0 += 16) {
            v2f b0 = *(const v2f*)(wp + k0);
            v2f b1 = *(const v2f*)(wp + k0 + 4);
            v2f b2 = *(const v2f*)(wp + k0 + 8);
            v2f b3 = *(const v2f*)(wp + k0 + 12);
            v2f a0 = *(const v2f*)(ap + k0);
            v2f a1 = *(const v2f*)(ap + k0 + 4);
            v2f a2 = *(const v2f*)(ap + k0 + 8);
            v2f a3 = *(const v2f*)(ap + k0 + 12);
            acc = __builtin_amdgcn_wmma_f32_16x16x4_f32(false, a0, false, b0, (short)0, acc, false, false);
            acc = __builtin_amdgcn_wmma_f32_16x16x4_f32(false, a1, false, b1, (short)0, acc, false, false);
            acc = __builtin_amdgcn_wmma_f32_16x16x4_f32(false, a2, false, b2, (short)0, acc, false, false);
            acc = __builtin_amdgcn_wmma_f32_16x16x4_f32(false, a3, false, b3, (short)0, acc, false, false);
        }
#pragma unroll
        for (int v = 0; v < 8; ++v) {
            int r = v + 8 * half;
            H[(size_t)(row0 + r) * HSTRIDE + col] = acc[v];
        }
    }
}

// ---------------------------------------------------------------------------
// Combine per-token log-probs per the reference's masked-select logic.
// ---------------------------------------------------------------------------
__global__ void combine_kernel(const int* __restrict__ y,
                               const float* __restrict__ head_lp,
                               const float* __restrict__ cl_lp,
                               const float* __restrict__ lp0,
                               const float* __restrict__ lp1,
                               float* __restrict__ out) {
    int n = blockIdx.x * blockDim.x + threadIdx.x;
    if (n >= NTOK) return;
    int yy = y[n];
    float o = head_lp[n];
    if (yy >= CUT0 && yy < CUT1) o = cl_lp[2 * n + 0] + lp0[n];
    else if (yy >= CUT1)         o = cl_lp[2 * n + 1] + lp1[n];
    out[n] = o;
}

__global__ void loss_kernel(const float* __restrict__ out, float* __restrict__ loss) {
    __shared__ float red[256];
    float sum = 0.f;
    for (int i = threadIdx.x; i < NTOK; i += 256) sum += out[i];
    red[threadIdx.x] = sum;
    __syncthreads();
    for (int st = 128; st > 0; st >>= 1) {
        if (threadIdx.x < st) red[threadIdx.x] += red[threadIdx.x + st];
        __syncthreads();
    }
    if (threadIdx.x == 0) *loss = -red[0] / (float)NTOK;
}

// ---------------------------------------------------------------------------
// Host launcher
// ---------------------------------------------------------------------------
extern "C" void kernel_launch(void* const* d_in, const int* in_sizes, int n_in,
                              void* d_out, int out_size, void* d_ws, size_t ws_size,
                              hipStream_t stream) {
    const float* x      = (const float*)d_in[0];   // [4096, 1024]
    const int*   y      = (const int*)  d_in[1];   // [4096]
    const float* head_w = (const float*)d_in[2];   // [1677, 1024]
    const float* head_b = (const float*)d_in[3];   // [1677]
    const float* proj0  = (const float*)d_in[4];   // [256, 1024]
    const float* w0     = (const float*)d_in[5];   // [3350, 256]
    const float* proj1  = (const float*)d_in[6];   // [64, 1024]
    const float* w1     = (const float*)d_in[7];   // [45232, 64]

    float* out  = (float*)d_out;                   // [4096] lp + [1] loss
    float* loss = out + NTOK;

    // workspace layout
    float* H         = (float*)d_ws;                     // 4096 x 320
    float* m_part    = H + (size_t)NTOK * HSTRIDE;       // MAXCHUNKS x 4096
    float* s_part    = m_part + MAXCHUNKS * NTOK;        // MAXCHUNKS x 4096
    float* head_lp   = s_part + MAXCHUNKS * NTOK;        // 4096
    float* cl_lp     = head_lp + NTOK;                   // 4096 x 2
    float* lp0       = cl_lp + 2 * NTOK;                 // 4096
    float* lp1       = lp0 + NTOK;                       // 4096
    float* tgt_logit = lp1 + NTOK;                       // 4096 (reused per stage)
    float* cl_logit  = tgt_logit + NTOK;                 // 4096 x 2
    int*   tgt_head  = (int*)(cl_logit + 2 * NTOK);      // 4096
    int*   tgt0      = tgt_head + NTOK;                  // 4096
    int*   tgt1      = tgt0 + NTOK;                      // 4096

    const dim3 blk128(128), blk256(256);
    const dim3 grid_tok(NTOK / 256);                     // 16
    const dim3 grid_gd(NTOK / 8);                        // 512 (gather-dot)

    prep_kernel<<<grid_tok, blk256, 0, stream>>>(y, tgt_head, tgt0, tgt1);

    // ---- head: K=1024, V=1677, bias; RT=2, 4 V-chunks ----
    {
        const int chunks = 4;
        const int ntiles = (HEAD_SIZE + 15) / 16;            // 105
        const int tpc = (ntiles + chunks - 1) / chunks;      // 27
        fused_lse_partial<2><<<dim3(NTOK / 32, chunks), blk128,
                               32 * (NH + 4) * sizeof(float), stream>>>(
            x, NH, 0, head_w, head_b, NH, HEAD_SIZE, tpc, m_part, s_part);
        gather_dot_kernel<<<grid_gd, blk256, 0, stream>>>(
            x, NH, 0, head_w, head_b, NH, tgt_head, -1, tgt_logit, 1, 0);
        gather_dot_kernel<<<grid_gd, blk256, 0, stream>>>(
            x, NH, 0, head_w, head_b, NH, nullptr, CUT0, cl_logit, 2, 0);
        gather_dot_kernel<<<grid_gd, blk256, 0, stream>>>(
            x, NH, 0, head_w, head_b, NH, nullptr, CUT0 + 1, cl_logit, 2, 1);
        lse_merge_kernel<<<grid_tok, blk256, 0, stream>>>(
            m_part, s_part, chunks, tgt_logit, cl_logit, head_lp, cl_lp);
    }

    // ---- projections: H = x @ [proj0; proj1]^T ----
    proj_kernel<<<dim3(NTOK / 16), blk128, 16 * (NH + 4) * sizeof(float), stream>>>(
        x, proj0, proj1, H);

    // ---- cluster 0: K=256 (cols 0..255 of H), V=3350; RT=4, 8 chunks ----
    {
        const int chunks = 8;
        const int ntiles = (SZ0 + 15) / 16;                  // 210
        const int tpc = (ntiles + chunks - 1) / chunks;      // 27
        fused_lse_partial<4><<<dim3(NTOK / 64, chunks), blk128,
                               64 * (D0 + 4) * sizeof(float), stream>>>(
            H, HSTRIDE, 0, w0, nullptr, D0, SZ0, tpc, m_part, s_part);
        gather_dot_kernel<<<grid_gd, blk256, 0, stream>>>(
            H, HSTRIDE, 0, w0, nullptr, D0, tgt0, -1, tgt_logit, 1, 0);
        lse_merge_kernel<<<grid_tok, blk256, 0, stream>>>(
            m_part, s_part, chunks, tgt_logit, nullptr, lp0, nullptr);
    }

    // ---- cluster 1: K=64 (cols 256..319 of H), V=45232; RT=4, 16 chunks ----
    {
        const int chunks = 16;
        const int ntiles = (SZ1 + 15) / 16;                  // 2827
        const int tpc = (ntiles + chunks - 1) / chunks;      // 177
        fused_lse_partial<4><<<dim3(NTOK / 64, chunks), blk128,
                               64 * (D1 + 4) * sizeof(float), stream>>>(
            H, HSTRIDE, D0, w1, nullptr, D1, SZ1, tpc, m_part, s_part);
        gather_dot_kernel<<<grid_gd, blk256, 0, stream>>>(
            H, HSTRIDE, D0, w1, nullptr, D1, tgt1, -1, tgt_logit, 1, 0);
        lse_merge_kernel<<<grid_tok, blk256, 0, stream>>>(
            m_part, s_part, chunks, tgt_logit, nullptr, lp1, nullptr);
    }

    combine_kernel<<<grid_tok, blk256, 0, stream>>>(y, head_lp, cl_lp, lp0, lp1, out);
    loss_kernel<<<dim3(1), blk256, 0, stream>>>(out, loss);
}